// MyHingeLoss_85426899517512
// MI455X (gfx1250) — compile-verified
//
#include <hip/hip_runtime.h>
#include <hip/hip_bf16.h>
#include <math.h>

// ---------------------------------------------------------------------------
// Cosine hinge loss, MI455X (gfx1250, wave32).
//   N = 32768 rows, D = 512, f32.
// Strategy: one wave per 16-row tile. Stream output/target once (128 MB total,
// HBM-bound at 23.3 TB/s). The same loaded registers feed:
//   - V_WMMA_F32_16X16X4_F32 accumulation of S = output @ t5^T (t5 in LDS)
//   - VALU FMAs for per-row dot / ||output|| / ||target||
// Per-block partials -> d_ws, deterministic tree reduce -> d_out.
// ---------------------------------------------------------------------------

typedef __attribute__((ext_vector_type(2))) float v2f;
typedef __attribute__((ext_vector_type(8))) float v8f;

#define HN 32768
#define HD 512
#define MARGIN 0.1f
#define HEPS 1e-6f
#define NCMP 5
#define WAVES_PER_BLOCK 8
#define ROWS_PER_BLOCK (WAVES_PER_BLOCK * 16)     // 128
#define NBLOCKS (HN / ROWS_PER_BLOCK)             // 256
#define T5_PAD 520                                // padded LDS stride (floats)

__global__ __launch_bounds__(256) void hinge_main_kernel(
    const float* __restrict__ output,
    const float* __restrict__ target,
    float* __restrict__ block_partials)
{
    __shared__ float lds_t5[NCMP * T5_PAD];       // 5 x 512 padded
    __shared__ __align__(8) float lds_zero[2];    // zero B-columns (cols 5..15)
    __shared__ float lds_tn5[NCMP];
    __shared__ float lds_wsum[WAVES_PER_BLOCK];

    const int tid  = threadIdx.x;
    const int lane = tid & 31;
    const int wid  = tid >> 5;

    // ---- stage t5 (first 5 target rows) into LDS, padded stride ----
    for (int idx = tid; idx < NCMP * HD; idx += 256) {
        int rr = idx >> 9;          // / 512
        int cc = idx & (HD - 1);    // % 512
        lds_t5[rr * T5_PAD + cc] = target[rr * HD + cc];
    }
    if (tid < 2) lds_zero[tid] = 0.0f;
    __syncthreads();

    // ---- tn5: norms of the 5 compare rows (redundant per block, cheap) ----
    if (tid < NCMP) {
        float s = 0.0f;
        const float* r = &lds_t5[tid * T5_PAD];
        for (int k = 0; k < HD; ++k) s += r[k] * r[k];
        lds_tn5[tid] = sqrtf(s);
    }
    __syncthreads();

    // ---- per-wave 16-row tile ----
    const int r0    = (blockIdx.x * WAVES_PER_BLOCK + wid) * 16;
    const int mrow  = lane & 15;        // row within tile (A) / column (B)
    const int khalf = (lane >> 4) * 2;  // K sub-pair: lanes 0-15 -> K%4 in {0,1}, 16-31 -> {2,3}

    const float* ap = output + (size_t)(r0 + mrow) * HD + khalf;
    const float* tp = target + (size_t)(r0 + mrow) * HD + khalf;
    const bool colValid = (mrow < NCMP);
    const float* bp = colValid ? &lds_t5[mrow * T5_PAD + khalf] : &lds_zero[0];
    const int bstep = colValid ? 4 : 0;

    v8f  acc = {};          // S-dot 16x16 tile accumulator (cols 5..15 are zero)
    float posd = 0.0f;      // dot(output[row], target[row]) partial (half of K)
    float on2  = 0.0f;      // ||output[row]||^2 partial
    float tn2  = 0.0f;      // ||target[row]||^2 partial

    #pragma unroll 4
    for (int s = 0; s < HD / 4; ++s) {
        v2f a = *(const v2f*)ap;    // global_load_b64
        v2f t = *(const v2f*)tp;    // global_load_b64
        v2f b = *(const v2f*)bp;    // ds_load_b64

        posd += a.x * t.x + a.y * t.y;
        on2  += a.x * a.x + a.y * a.y;
        tn2  += t.x * t.x + t.y * t.y;

        // D = A(16x4 f32) x B(4x16 f32) + C   -> v_wmma_f32_16x16x4_f32
        acc = __builtin_amdgcn_wmma_f32_16x16x4_f32(
                  /*neg_a=*/false, a, /*neg_b=*/false, b,
                  /*c_mod=*/(short)0, acc,
                  /*reuse_a=*/false, /*reuse_b=*/false);

        ap += 4; tp += 4; bp += bstep;
    }

    // ---- combine complementary K-halves (lane L <-> L^16 hold same row) ----
    posd += __shfl_xor(posd, 16, 32);
    on2  += __shfl_xor(on2, 16, 32);
    tn2  += __shfl_xor(tn2, 16, 32);

    const float on_v  = sqrtf(on2);
    const float tn_v  = sqrtf(tn2);
    const float pos_n = posd / fmaxf(on_v * tn_v, HEPS);

    // ---- hinge over this wave's 16x16 S tile ----
    // C/D layout: lane L, vgpr v -> M = v + 8*(L>>4), Ncol = L%16
    const int   ncol  = mrow;
    const float tn5v  = colValid ? lds_tn5[ncol] : 1.0f;
    const int   mbase = (lane >> 4) * 8;

    float hsum = 0.0f;
    #pragma unroll
    for (int v = 0; v < 8; ++v) {
        int   M     = mbase + v;
        float posnM = __shfl(pos_n, M, 32);   // lane M holds row M's pos
        float onM   = __shfl(on_v, M, 32);
        float sval  = acc[v] / fmaxf(onM * tn5v, HEPS);
        float h     = fmaxf(0.0f, MARGIN - posnM + sval);
        if (!colValid) h = 0.0f;              // dead columns 5..15
        if ((r0 + M) == ncol) h = 0.0f;       // diagonal mask i==j (j<5)
        hsum += h;
    }

    // ---- wave reduce, then block reduce ----
    #pragma unroll
    for (int off = 16; off > 0; off >>= 1)
        hsum += __shfl_xor(hsum, off, 32);
    if (lane == 0) lds_wsum[wid] = hsum;
    __syncthreads();
    if (tid == 0) {
        float s = 0.0f;
        #pragma unroll
        for (int w = 0; w < WAVES_PER_BLOCK; ++w) s += lds_wsum[w];
        block_partials[blockIdx.x] = s;
    }
}

__global__ __launch_bounds__(256) void hinge_finalize_kernel(
    const float* __restrict__ block_partials,
    float* __restrict__ out)
{
    __shared__ float sred[256];
    const int tid = threadIdx.x;
    sred[tid] = (tid < NBLOCKS) ? block_partials[tid] : 0.0f;
    __syncthreads();
    #pragma unroll
    for (int off = 128; off > 0; off >>= 1) {
        if (tid < off) sred[tid] += sred[tid + off];
        __syncthreads();
    }
    if (tid == 0) {
        const float inv_count = 1.0f / (float)(HN * NCMP - NCMP);
        out[0] = sred[0] * inv_count;
    }
}

extern "C" void kernel_launch(void* const* d_in, const int* in_sizes, int n_in,
                              void* d_out, int out_size, void* d_ws, size_t ws_size,
                              hipStream_t stream) {
    const float* output = (const float*)d_in[0];
    const float* target = (const float*)d_in[1];
    float* out = (float*)d_out;
    float* ws  = (float*)d_ws;   // NBLOCKS floats of partials

    hinge_main_kernel<<<NBLOCKS, 256, 0, stream>>>(output, target, ws);
    hinge_finalize_kernel<<<1, 256, 0, stream>>>(ws, out);
}